// ResponseGenerationNetwork_11115375362787
// MI455X (gfx1250) — compile-verified
//
#include <hip/hip_runtime.h>
#include <hip/hip_bf16.h>
#include <math.h>

// Problem constants (match reference; max_length scalar is fixed at 100).
#define BB   16
#define HH   512
#define LL   3
#define TT   100
#define VV   32000
#define G4H  2048           // 4*H
#define NWG_LSTM 32         // H / 16 hidden blocks
#define KCHUNK 128          // 1024 total K split over 8 waves
#define MT   4              // M (time) blocking in projection

typedef __attribute__((ext_vector_type(2))) float v2f;
typedef __attribute__((ext_vector_type(8))) float v8f;

__device__ __forceinline__ float sigmf(float x) { return 1.0f / (1.0f + expf(-x)); }

// ---------------------------------------------------------------------------
// Device-scope grid barrier among NWG_LSTM co-resident workgroups.
// ---------------------------------------------------------------------------
__device__ __forceinline__ void grid_barrier(int* cnt, int* gen, int nwg) {
  __syncthreads();
  if (threadIdx.x == 0) {
    __threadfence();
    int g = __hip_atomic_load(gen, __ATOMIC_RELAXED, __HIP_MEMORY_SCOPE_AGENT);
    int arrived = __hip_atomic_fetch_add(cnt, 1, __ATOMIC_ACQ_REL,
                                         __HIP_MEMORY_SCOPE_AGENT) + 1;
    if (arrived == nwg) {
      __hip_atomic_store(cnt, 0, __ATOMIC_RELAXED, __HIP_MEMORY_SCOPE_AGENT);
      __hip_atomic_fetch_add(gen, 1, __ATOMIC_ACQ_REL, __HIP_MEMORY_SCOPE_AGENT);
    } else {
      while (__hip_atomic_load(gen, __ATOMIC_ACQUIRE,
                               __HIP_MEMORY_SCOPE_AGENT) == g) {
        __builtin_amdgcn_s_sleep(1);
      }
    }
    __threadfence();
  }
  __syncthreads();
}

// ---------------------------------------------------------------------------
// Init: x0 = style_table[style_id], h = broadcast(context), c = 0, barrier = 0
// ---------------------------------------------------------------------------
__global__ void init_kernel(const float* __restrict__ ctx,
                            const int* __restrict__ sid,
                            const float* __restrict__ stab,
                            float* __restrict__ x0,
                            float* __restrict__ h_state,
                            float* __restrict__ c_state,
                            int* __restrict__ bar) {
  int i = blockIdx.x * blockDim.x + threadIdx.x;
  if (i < BB * HH) {
    int b = i / HH, h = i % HH;
    x0[i] = stab[sid[b] * HH + h];
  }
  if (i < LL * BB * HH) {
    h_state[i] = ctx[i % (BB * HH)];
    c_state[i] = 0.0f;
  }
  if (i < 2) bar[i] = 0;
}

// ---------------------------------------------------------------------------
// Persistent LSTM kernel: 32 WGs x 256 threads. WG `wg` owns hidden columns
// [wg*16, wg*16+16). Waves 0-3 accumulate x @ W_ih^T over K chunks of 128,
// waves 4-7 accumulate h @ W_hh^T. Partials reduced via LDS; wave 0 applies
// gate nonlinearities and updates (h, c). Grid barrier per layer.
// Inner K loop is software-pipelined (1 deep) so fragment loads for step k+4
// are in flight while the four WMMAs of step k execute.
// ---------------------------------------------------------------------------
__global__ __launch_bounds__(256) void lstm_kernel(
    const float* __restrict__ W_ih, const float* __restrict__ W_hh,
    const float* __restrict__ b_ih, const float* __restrict__ b_hh,
    const float* __restrict__ x0,
    float* __restrict__ h_state, float* __restrict__ c_state,
    float* __restrict__ h_seq,
    int* __restrict__ bar_cnt, int* __restrict__ bar_gen) {
  __shared__ float red[8][4][8][32];   // 32 KB: [wave][gate][vgpr][lane]

  const int wg   = blockIdx.x;                                   // hidden block
  const int lane = threadIdx.x & 31;
  // Wave index is uniform within the wave: force it into an SGPR so the K
  // loop bounds and operand selection are scalar (no EXEC-masked looping).
  const int wave = __builtin_amdgcn_readfirstlane(threadIdx.x >> 5);
  const int n    = lane & 15;          // tile column (hidden within block)
  const int koff = (lane >> 4) << 1;   // 0 or 2: f32 WMMA A/B K sub-offset
  const int j    = wg * 16 + n;        // global hidden index for this column
  const int m    = lane & 15;          // A-fragment row (batch)

  const int k0   = (wave & 3) * KCHUNK;
  const int kend = k0 + KCHUNK;
  const bool lowhalf = (wave < 4);

  for (int t = 0; t < TT; ++t) {
    for (int l = 0; l < LL; ++l) {
      const float* x = (l == 0)
          ? ((t == 0) ? x0 : h_seq + (size_t)(t - 1) * BB * HH)
          : h_state + (size_t)(l - 1) * BB * HH;
      const float* hprev = h_state + (size_t)l * BB * HH;

      const float* A  = lowhalf ? x : hprev;
      const float* Wm = (lowhalf ? W_ih : W_hh) + (size_t)l * G4H * HH;

      const float* arow = A + (size_t)m * HH + koff;
      const float* wrow[4];
#pragma unroll
      for (int g = 0; g < 4; ++g)
        wrow[g] = Wm + (size_t)(g * HH + wg * 16 + n) * HH + koff;

      v8f acc[4] = {};                 // one 16x16 f32 tile per gate

      // ---- software-pipelined K loop (prefetch next fragments) ----
      v2f a_cur = *(const v2f*)(arow + k0);
      v2f b_cur[4];
#pragma unroll
      for (int g = 0; g < 4; ++g) b_cur[g] = *(const v2f*)(wrow[g] + k0);

      for (int k = k0; k < kend; k += 4) {
        int kn = k + 4;
        kn = (kn < kend) ? kn : k0;    // wrap on last iter (dummy, in-bounds)
        v2f a_nxt = *(const v2f*)(arow + kn);
        v2f b_nxt[4];
#pragma unroll
        for (int g = 0; g < 4; ++g) b_nxt[g] = *(const v2f*)(wrow[g] + kn);

#pragma unroll
        for (int g = 0; g < 4; ++g)
          acc[g] = __builtin_amdgcn_wmma_f32_16x16x4_f32(
              false, a_cur, false, b_cur[g], (short)0, acc[g], false, false);

        a_cur = a_nxt;
#pragma unroll
        for (int g = 0; g < 4; ++g) b_cur[g] = b_nxt[g];
      }

#pragma unroll
      for (int g = 0; g < 4; ++g)
#pragma unroll
        for (int r = 0; r < 8; ++r) red[wave][g][r][lane] = acc[g][r];
      __syncthreads();

      if (wave == 0) {
        float bsum[4];
#pragma unroll
        for (int g = 0; g < 4; ++g)
          bsum[g] = b_ih[l * G4H + g * HH + j] + b_hh[l * G4H + g * HH + j];
#pragma unroll
        for (int r = 0; r < 8; ++r) {
          float s0 = bsum[0], s1 = bsum[1], s2 = bsum[2], s3 = bsum[3];
#pragma unroll
          for (int w = 0; w < 8; ++w) {
            s0 += red[w][0][r][lane];
            s1 += red[w][1][r][lane];
            s2 += red[w][2][r][lane];
            s3 += red[w][3][r][lane];
          }
          const int mrow = r + ((lane >> 4) << 3);   // batch row (C/D layout)
          const size_t idx = (size_t)l * BB * HH + (size_t)mrow * HH + j;
          float cprev = c_state[idx];
          float ig = sigmf(s0), fg = sigmf(s1), gg = tanhf(s2), og = sigmf(s3);
          float cn = fg * cprev + ig * gg;
          float hn = og * tanhf(cn);
          c_state[idx] = cn;
          h_state[idx] = hn;
          if (l == LL - 1)
            h_seq[(size_t)t * BB * HH + (size_t)mrow * HH + j] = hn;
        }
      }
      __syncthreads();
      grid_barrier(bar_cnt, bar_gen, NWG_LSTM);
    }
  }
}

// ---------------------------------------------------------------------------
// Projection: logits[t][b][v] = h_seq[t][b] . W_out[v] + b_out[v], stored
// transposed as out[b][t][0][v]. One wave per (t-group of 4, 16-wide v block),
// software-pipelined K loop, W_out fragment shared across the 4 time tiles.
// ---------------------------------------------------------------------------
__global__ __launch_bounds__(256) void proj_kernel(
    const float* __restrict__ h_seq, const float* __restrict__ W_out,
    const float* __restrict__ b_out, float* __restrict__ out) {
  const int lane = threadIdx.x & 31;
  const int wave = __builtin_amdgcn_readfirstlane(threadIdx.x >> 5);
  const int NB = VV / 16;                       // 2000 vocab blocks
  const int id = blockIdx.x * 8 + wave;         // 0 .. (TT/MT)*NB - 1
  const int tg = id / NB;                       // time group (4 steps each)
  const int nb = id % NB;                       // vocab block

  const int n    = lane & 15;
  const int koff = (lane >> 4) << 1;
  const int m    = lane & 15;                   // batch row for A fragment

  v8f acc[MT] = {};
  const float* wrow = W_out + (size_t)(nb * 16 + n) * HH + koff;
  const float* arow[MT];
#pragma unroll
  for (int mt = 0; mt < MT; ++mt)
    arow[mt] = h_seq + ((size_t)(tg * MT + mt) * BB + m) * HH + koff;

  // ---- software-pipelined K loop ----
  v2f b_cur = *(const v2f*)(wrow);
  v2f a_cur[MT];
#pragma unroll
  for (int mt = 0; mt < MT; ++mt) a_cur[mt] = *(const v2f*)(arow[mt]);

  for (int k = 0; k < HH; k += 4) {
    int kn = k + 4;
    kn = (kn < HH) ? kn : 0;                    // wrap on last iter
    v2f b_nxt = *(const v2f*)(wrow + kn);
    v2f a_nxt[MT];
#pragma unroll
    for (int mt = 0; mt < MT; ++mt) a_nxt[mt] = *(const v2f*)(arow[mt] + kn);

#pragma unroll
    for (int mt = 0; mt < MT; ++mt)
      acc[mt] = __builtin_amdgcn_wmma_f32_16x16x4_f32(
          false, a_cur[mt], false, b_cur, (short)0, acc[mt], false, false);

    b_cur = b_nxt;
#pragma unroll
    for (int mt = 0; mt < MT; ++mt) a_cur[mt] = a_nxt[mt];
  }

  const float bias = b_out[nb * 16 + n];
  const int mbase = (lane >> 4) << 3;           // 0 or 8
#pragma unroll
  for (int mt = 0; mt < MT; ++mt) {
    const int t = tg * MT + mt;
#pragma unroll
    for (int r = 0; r < 8; ++r) {
      const int b = mbase + r;                  // batch index
      out[((size_t)b * TT + t) * VV + nb * 16 + n] = acc[mt][r] + bias;
    }
  }
}

// ---------------------------------------------------------------------------
extern "C" void kernel_launch(void* const* d_in, const int* in_sizes, int n_in,
                              void* d_out, int out_size, void* d_ws, size_t ws_size,
                              hipStream_t stream) {
  const float* context     = (const float*)d_in[0];   // [B,H]
  const int*   style_id    = (const int*)  d_in[1];   // [B]
  const float* style_table = (const float*)d_in[2];   // [5,H]
  const float* W_ih        = (const float*)d_in[3];   // [L,4H,H]
  const float* W_hh        = (const float*)d_in[4];   // [L,4H,H]
  const float* b_ih        = (const float*)d_in[5];   // [L,4H]
  const float* b_hh        = (const float*)d_in[6];   // [L,4H]
  const float* W_out       = (const float*)d_in[7];   // [V,H]
  const float* b_out       = (const float*)d_in[8];   // [V]
  // d_in[9] = max_length (==100, hardcoded as TT)
  float* out = (float*)d_out;

  // Workspace layout (floats)
  float* ws      = (float*)d_ws;
  float* x0      = ws;                                  // B*H
  float* h_state = x0 + BB * HH;                        // L*B*H
  float* c_state = h_state + (size_t)LL * BB * HH;      // L*B*H
  float* h_seq   = c_state + (size_t)LL * BB * HH;      // T*B*H
  int*   bar     = (int*)(h_seq + (size_t)TT * BB * HH);

  {
    int nelem = LL * BB * HH;
    int blocks = (nelem + 255) / 256;
    init_kernel<<<blocks, 256, 0, stream>>>(context, style_id, style_table,
                                            x0, h_state, c_state, bar);
  }
  lstm_kernel<<<NWG_LSTM, 256, 0, stream>>>(W_ih, W_hh, b_ih, b_hh, x0,
                                            h_state, c_state, h_seq,
                                            bar, bar + 1);
  {
    int nwaves = (TT / MT) * (VV / 16);   // 25 * 2000 = 50000
    int blocks = nwaves / 8;              // 6250
    proj_kernel<<<blocks, 256, 0, stream>>>(h_seq, W_out, b_out, out);
  }
}